// GCNLayer_22101901705838
// MI455X (gfx1250) — compile-verified
//
#include <hip/hip_runtime.h>
#include <hip/hip_bf16.h>

typedef __attribute__((ext_vector_type(2))) float v2f;
typedef __attribute__((ext_vector_type(8))) float v8f;

#define F 128          // F_IN == F_OUT == 128
#define NNODES 50000
#define WAVE 32

// ---------------------------------------------------------------------------
// Kernel 1: zero the aggregation workspace (float4-vectorized)
// ---------------------------------------------------------------------------
__global__ void gcn_zero_kernel(float4* __restrict__ p, int n4) {
    int i = blockIdx.x * blockDim.x + threadIdx.x;
    if (i < n4) p[i] = make_float4(0.f, 0.f, 0.f, 0.f);
}

// ---------------------------------------------------------------------------
// Kernel 2: edge scatter. One wave32 per edge; lane l handles feature columns
// [4l, 4l+4). Source row load is one coalesced 512B burst per wave; the
// scatter is 4 relaxed agent-scope f32 atomics per lane into agg (L2-resident,
// 25.6 MB << 192 MB L2).
// ---------------------------------------------------------------------------
__global__ void gcn_edge_scatter_kernel(const float* __restrict__ feature,
                                        const float* __restrict__ weight,
                                        const int* __restrict__ src,
                                        const int* __restrict__ dst,
                                        float* __restrict__ agg,
                                        int E) {
    int gtid = blockIdx.x * blockDim.x + threadIdx.x;
    int e    = gtid >> 5;          // wave index == edge index (wave32)
    int lane = gtid & (WAVE - 1);
    if (e >= E) return;

    int   s = src[e];
    int   d = dst[e];
    float w = weight[e] + 1.0f;

    const float4* frow = (const float4*)(feature + (size_t)s * F);
    float4 v = frow[lane];

    float* arow = agg + (size_t)d * F + lane * 4;
    __hip_atomic_fetch_add(arow + 0, w * v.x, __ATOMIC_RELAXED, __HIP_MEMORY_SCOPE_AGENT);
    __hip_atomic_fetch_add(arow + 1, w * v.y, __ATOMIC_RELAXED, __HIP_MEMORY_SCOPE_AGENT);
    __hip_atomic_fetch_add(arow + 2, w * v.z, __ATOMIC_RELAXED, __HIP_MEMORY_SCOPE_AGENT);
    __hip_atomic_fetch_add(arow + 3, w * v.w, __ATOMIC_RELAXED, __HIP_MEMORY_SCOPE_AGENT);
}

// ---------------------------------------------------------------------------
// Kernel 3: fused epilogue + GEMM via V_WMMA_F32_16X16X4_F32 (fp32 WMMA).
//   out[m][n] = sum_k (feature[m][k]*(sw[m]+1) + agg[m][k]) * W[n][k] + b[n]
// Block = 256 threads = 8 waves. Block handles 16 rows (one M tile);
// wave j handles output columns [16j, 16j+16). K=128 -> 32 wmma steps.
//
// ISA f32 operand layouts (CDNA5 ISA):
//   A 16x4 : lane holds row M = lane%16; VGPR0/1 = K = 2*(lane/16) + {0,1}
//   B 4x16 : lane holds col N = lane%16; VGPR0/1 = K = 2*(lane/16) + {0,1}
//   C/D    : VGPR v -> row M = v + 8*(lane/16), col N = lane%16
// No divergence anywhere (EXEC must be all ones for WMMA).
// ---------------------------------------------------------------------------
__global__ void gcn_wmma_epilogue_kernel(const float* __restrict__ feature,
                                         const float* __restrict__ self_weight,
                                         const float* __restrict__ agg,
                                         const float* __restrict__ Wm,
                                         const float* __restrict__ bias,
                                         float* __restrict__ out) {
    const int row0 = blockIdx.x * 16;          // M tile base (3125*16 == 50000)
    const int wv   = threadIdx.x >> 5;         // 0..7 -> N tile
    const int lane = threadIdx.x & (WAVE - 1);
    const int n0   = wv * 16;
    const int m    = lane & 15;                // row (A) / col (B) within tile
    const int half = lane >> 4;                // K-pair selector

    const int    row  = row0 + m;
    const float  sw1  = self_weight[row] + 1.0f;
    const float* frow = feature + (size_t)row * F;
    const float* arow = agg     + (size_t)row * F;
    const float* wrow = Wm      + (size_t)(n0 + m) * F;   // B[k][n] = W[n][k]

    v8f c = {};
#pragma unroll
    for (int kk = 0; kk < F / 4; ++kk) {
        const int k = kk * 4 + half * 2;
        v2f a, b;
        a.x = fmaf(frow[k],     sw1, arow[k]);
        a.y = fmaf(frow[k + 1], sw1, arow[k + 1]);
        b.x = wrow[k];
        b.y = wrow[k + 1];
        c = __builtin_amdgcn_wmma_f32_16x16x4_f32(
                /*neg_a=*/false, a, /*neg_b=*/false, b,
                /*c_mod=*/(short)0, c, /*reuse_a=*/false, /*reuse_b=*/false);
    }

    const float bn = bias[n0 + m];
#pragma unroll
    for (int v = 0; v < 8; ++v) {
        out[(size_t)(row0 + v + 8 * half) * F + (n0 + m)] = c[v] + bn;
    }
}

// ---------------------------------------------------------------------------
// Launcher
// ---------------------------------------------------------------------------
extern "C" void kernel_launch(void* const* d_in, const int* in_sizes, int n_in,
                              void* d_out, int out_size, void* d_ws, size_t ws_size,
                              hipStream_t stream) {
    const float* feature     = (const float*)d_in[0];   // [N,128]
    const float* self_weight = (const float*)d_in[1];   // [N,1]
    const float* weight      = (const float*)d_in[2];   // [E]
    const int*   src         = (const int*)d_in[3];     // [E]
    const int*   dst         = (const int*)d_in[4];     // [E]
    const float* Wm          = (const float*)d_in[5];   // [128,128]
    const float* bias        = (const float*)d_in[6];   // [128]
    float*       out         = (float*)d_out;           // [N,128]

    const int N = in_sizes[1];                 // 50000 (self_weight is N*1)
    const int E = in_sizes[2];                 // 800000

    float* agg = (float*)d_ws;                 // N*128 floats = 25.6 MB scratch

    // 1) zero agg
    {
        int n4 = (N * F) / 4;
        int threads = 256;
        int blocks = (n4 + threads - 1) / threads;
        gcn_zero_kernel<<<blocks, threads, 0, stream>>>((float4*)agg, n4);
    }

    // 2) edge scatter: one wave32 per edge
    {
        int threads = 256;                      // 8 waves per block
        long long total_threads = (long long)E * WAVE;
        int blocks = (int)((total_threads + threads - 1) / threads);
        gcn_edge_scatter_kernel<<<blocks, threads, 0, stream>>>(
            feature, weight, src, dst, agg, E);
    }

    // 3) fused self-loop + GEMM (WMMA f32)
    {
        int threads = 256;                      // 8 waves, one N-tile each
        int blocks = N / 16;                    // 3125, exact
        gcn_wmma_epilogue_kernel<<<blocks, threads, 0, stream>>>(
            feature, self_weight, agg, Wm, bias, out);
    }
}